// AdaptivePoolOrGaussian_48644799594971
// MI455X (gfx1250) — compile-verified
//
#include <hip/hip_runtime.h>

// ---------------------------------------------------------------------------
// AdaptivePoolOrGaussian on MI455X (gfx1250)
//
// All 8 smoothers (5 avg-pools, 3 Gaussians) are separable symmetric filters
// g_i (x) g_i with half-width <= 5.  The softmax-weighted sum is linear, so the
// whole op is ONE depthwise 11x11 convolution with K = sum_i w_i * g_i g_i^T,
// applied to 512 independent 256x256 fp32 planes.
//
// Memory floor: 256 MiB @ 23.3 TB/s ~ 11.5 us.  121 MAC/elem on fp32 VALU is
// ~4x slower than that, so the MACs go on the matrix pipe using the fp32 WMMA
// (V_WMMA_F32_16X16X4_F32, fp32 in / fp32 accumulate -> reference-precision).
// Per 16x16 output tile:  out = sum_{dr=0..10}  X_dr(16x28) * B_dr(28x16)
// where X_dr = input rows shifted by dr (from an LDS-staged 26x28 halo tile)
// and B_dr is a banded Toeplitz matrix built from combined-kernel row dr.
// 11 rows x 7 K-steps (K=28, 4 per WMMA) = 77 WMMAs per tile.
// ---------------------------------------------------------------------------

typedef __attribute__((ext_vector_type(2))) float v2f;
typedef __attribute__((ext_vector_type(8))) float v8f;

#define KS     11            // combined kernel size (2*5+1)
#define KHALF  5
#define NF     8             // number of blended filters
#define KSTEPS 7             // K = 28 -> 7 steps of 4
#define TW     28            // staged tile width  (16 + 2*5, padded to 28)
#define TH     26            // staged tile height (16 + 2*5)
#define IMG    256
#define BOPS_FLOATS (KS * KSTEPS * 32 * 2)   // [dr][s][lane][q] = 4928 floats

// ---------------------------------------------------------------------------
// Prep kernel: softmax weights, 1-D kernels, combined 11x11 kernel, and the
// Toeplitz B-operands pre-swizzled into WMMA-B lane layout:
//   B-operand VGPR q, lane L holds B_dr[k][n],  k = 4*s + 2*(L>>4) + q,
//   n = L & 15,  B_dr[k][n] = Kc[dr][k-n] if 0 <= k-n <= 10 else 0.
// ---------------------------------------------------------------------------
__global__ void AdaptivePoolOrGaussian_prep(const float* __restrict__ sigmas,
                                            const float* __restrict__ alpha,
                                            float* __restrict__ bops) {
    __shared__ float Kc[KS * KS];
    if (threadIdx.x == 0) {
        // softmax over the 8 blend logits
        float w[NF];
        float mx = alpha[0];
        for (int i = 1; i < NF; ++i) mx = fmaxf(mx, alpha[i]);
        float sum = 0.f;
        for (int i = 0; i < NF; ++i) { w[i] = expf(alpha[i] - mx); sum += w[i]; }
        for (int i = 0; i < NF; ++i) w[i] /= sum;

        // 1-D kernels, all centered, 11 taps
        float g[NF][KS];
        for (int i = 0; i < NF; ++i)
            for (int t = 0; t < KS; ++t) g[i][t] = 0.f;
        const int pk[5] = {0, 1, 2, 3, 5};          // pool half-widths
        for (int p = 0; p < 5; ++p) {
            const int k = pk[p];
            const float inv = 1.f / (float)(2 * k + 1);
            for (int t = KHALF - k; t <= KHALF + k; ++t) g[p][t] = inv;
        }
        for (int j = 0; j < 3; ++j) {               // Gaussians
            const float s = fabsf(sigmas[j]) + 1e-6f;
            float gs = 0.f;
            for (int t = 0; t < KS; ++t) {
                const float ax = (float)t - (float)KHALF;
                const float v  = expf(-0.5f * (ax / s) * (ax / s));
                g[5 + j][t] = v; gs += v;
            }
            for (int t = 0; t < KS; ++t) g[5 + j][t] /= gs;
        }
        // combined 2-D kernel  Kc = sum_i w_i * g_i (x) g_i
        for (int r = 0; r < KS; ++r)
            for (int c = 0; c < KS; ++c) {
                float a = 0.f;
                for (int i = 0; i < NF; ++i) a += w[i] * g[i][r] * g[i][c];
                Kc[r * KS + c] = a;
            }
    }
    __syncthreads();

    // Emit B operands in final lane layout.
    for (int e = threadIdx.x; e < BOPS_FLOATS; e += blockDim.x) {
        const int q    = e & 1;
        const int lane = (e >> 1) & 31;
        const int s    = (e >> 6) % KSTEPS;
        const int dr   = e / (KSTEPS * 64);
        const int k    = 4 * s + 2 * (lane >> 4) + q;
        const int n    = lane & 15;
        const int d    = k - n;                      // = dc + 5
        bops[e] = (d >= 0 && d < KS) ? Kc[dr * KS + d] : 0.f;
    }
}

// ---------------------------------------------------------------------------
// Main kernel: 256 threads = 8 waves, one 16x16 output tile per wave.
// ---------------------------------------------------------------------------
__global__ __launch_bounds__(256) void AdaptivePoolOrGaussian_conv(
        const float* __restrict__ x,
        const float* __restrict__ bops,
        float* __restrict__ out) {
    __shared__ float sB[BOPS_FLOATS];       // Toeplitz operands (19.7 KB)
    __shared__ float sT[8][TH * TW];        // per-wave halo tiles (23.3 KB)

    const int tid = threadIdx.x;
    for (int e = tid; e < BOPS_FLOATS; e += 256) sB[e] = bops[e];

    const int wave = tid >> 5;
    const int lane = tid & 31;
    const int gid  = blockIdx.x * 8 + wave;          // tile id, 512*256 total
    const int plane = gid >> 8;                      // 8*64 = 512 planes
    const int R = ((gid >> 4) & 15) << 4;            // tile row origin
    const int C = (gid & 15) << 4;                   // tile col origin

    const float* __restrict__ xp = x + (size_t)plane * (IMG * IMG);
    float* sTile = sT[wave];

    // Stage zero-padded 26x28 halo tile (rows R-5..R+20, cols C-5..C+22).
    #pragma unroll
    for (int it = 0; it < (TH * TW + 31) / 32; ++it) {
        const int idx = it * 32 + lane;
        if (idx < TH * TW) {
            const int r  = idx / TW;
            const int c  = idx - r * TW;
            const int gr = R - KHALF + r;
            const int gc = C - KHALF + c;
            float v = 0.f;
            if (gr >= 0 && gr < IMG && gc >= 0 && gc < IMG)
                v = xp[gr * IMG + gc];
            sTile[idx] = v;
        }
    }
    __syncthreads();   // sB ready for all waves; each wave's tile complete

    // A-operand (16x4 fp32) lane layout: lane L, VGPR q -> A[L%16][4s+2*(L>>4)+q]
    const int m     = lane & 15;
    const int half2 = (lane >> 4) << 1;                  // 0 or 2
    const float* aBase = sTile + m * TW + half2;         // 8B-aligned
    const float* bBase = sB + lane * 2;                  // 8B-aligned

    v8f acc = {};
    #pragma unroll
    for (int dr = 0; dr < KS; ++dr) {
        #pragma unroll
        for (int s = 0; s < KSTEPS; ++s) {
            const v2f a = *(const v2f*)(aBase + dr * TW + 4 * s);
            const v2f b = *(const v2f*)(bBase + (dr * KSTEPS + s) * 64);
            // D = A x B + C, fp32 everywhere (reference precision)
            acc = __builtin_amdgcn_wmma_f32_16x16x4_f32(
                      /*neg_a=*/false, a, /*neg_b=*/false, b,
                      /*c_mod=*/(short)0, acc,
                      /*reuse_a=*/false, /*reuse_b=*/false);
        }
    }

    // C/D layout: VGPR v -> row (v + 8*(lane>=16)), col = lane%16.
    float* __restrict__ op = out + (size_t)plane * (IMG * IMG);
    const int col   = C + m;
    const int rbase = R + (lane >> 4) * 8;
    #pragma unroll
    for (int v = 0; v < 8; ++v)
        op[(rbase + v) * IMG + col] = acc[v];
}

// ---------------------------------------------------------------------------
extern "C" void kernel_launch(void* const* d_in, const int* in_sizes, int n_in,
                              void* d_out, int out_size, void* d_ws, size_t ws_size,
                              hipStream_t stream) {
    const float* x      = (const float*)d_in[0];   // (8,64,256,256) fp32
    const float* sigmas = (const float*)d_in[1];   // (3,)
    const float* alpha  = (const float*)d_in[2];   // (8,)
    float*       bops   = (float*)d_ws;            // 4928 floats scratch
    float*       out    = (float*)d_out;

    AdaptivePoolOrGaussian_prep<<<1, 64, 0, stream>>>(sigmas, alpha, bops);

    // 512 planes * 16*16 tiles = 131072 tiles, 8 waves (tiles) per block.
    AdaptivePoolOrGaussian_conv<<<131072 / 8, 256, 0, stream>>>(x, bops, out);
}